// Samples_Loss_33191507263735
// MI455X (gfx1250) — compile-verified
//
#include <hip/hip_runtime.h>
#include <hip/hip_bf16.h>

// CDNA5 / gfx1250 Sinkhorn (geomloss SamplesLoss('sinkhorn', debias=False))
// B=8, N=M=2048, D=2. Cost matrix recomputed on the fly from Gram tiles via
// V_WMMA_F32_16X16X4_F32 (K padded 2->4; A lanes 16-31 zeroed so B needs no
// masking). Base-2 LSE: one v_exp_f32 per element on the trans pipe,
// co-executing with the matrix pipe. Zero HBM traffic for the 134MB C matrix.

typedef __attribute__((ext_vector_type(2))) float v2f;
typedef __attribute__((ext_vector_type(8))) float v8f;

#define NPTS 2048
#define NB   8
#define LOG2E 1.44269504088896340736f
#define LN2   0.69314718055994530942f

// ds_swizzle xor patterns within 32 lanes: j = ((i&0x1f)|0) ^ xor
#define SWZ_XOR1 0x041F
#define SWZ_XOR2 0x081F
#define SWZ_XOR4 0x101F
#define SWZ_XOR8 0x201F
#define XSWZ(v, imm) \
    __int_as_float(__builtin_amdgcn_ds_swizzle(__float_as_int(v), (imm)))

// ---------------------------------------------------------------- prep ----
__global__ void prep_kernel(const float* __restrict__ a,
                            const float* __restrict__ x,
                            const float* __restrict__ b,
                            const float* __restrict__ y,
                            float* __restrict__ la, float* __restrict__ lb,
                            float* __restrict__ sx, float* __restrict__ sy,
                            int n) {
    int i = blockIdx.x * blockDim.x + threadIdx.x;
    if (i >= n) return;
    la[i] = logf(a[i]);
    lb[i] = logf(b[i]);
    float x0 = x[2 * i], x1 = x[2 * i + 1];
    sx[i] = 0.5f * (x0 * x0 + x1 * x1);
    float y0 = y[2 * i], y1 = y[2 * i + 1];
    sy[i] = 0.5f * (y0 * y0 + y1 * y1);
}

// ------------------------------------------------------------- softmin ----
// out[b,i] = -eps * LSE_j( logw[b,j] + pot[b,j]*potScale - C(b,i,j)/eps )
// C = sqr_i + sqc_j - <P_i,Q_j>.  Base-2 form: arg2 = G*ie2 + hy2_j with
// ie2 = log2e/eps, hy2 = log2e*(logw + pot*ps - sqc/eps).
//   out = -eps*ln2*(M2 + log2 S) + sqr_i
// grid = NB*128 blocks x 128 threads. Block owns one 16-row tile; each of
// its 4 waves covers 512 columns; LDS combines the partial (max, sum) pairs.
__global__ __launch_bounds__(128) void softmin_kernel(
    const float* __restrict__ P,    // row points  [B,NPTS,2]
    const float* __restrict__ Q,    // col points  [B,NPTS,2]
    const float* __restrict__ logw, // [B,NPTS] col log-weights
    const float* __restrict__ pot,  // [B,NPTS] col potential
    const float* __restrict__ sqr,  // [B,NPTS] 0.5|P|^2
    const float* __restrict__ sqc,  // [B,NPTS] 0.5|Q|^2
    float* __restrict__ out,        // [B,NPTS]
    float eps, float potScale) {
    __shared__ float2 s_q[NPTS];    // 16 KB
    __shared__ float  s_hy[NPTS];   //  8 KB
    __shared__ float  red_m[4][16];
    __shared__ float  red_s[4][16];

    const float inv_eps = 1.0f / eps;
    const float ie2     = inv_eps * LOG2E;
    const int   b       = blockIdx.x >> 7;   // batch
    const int   rt      = blockIdx.x & 127;  // 16-row tile
    const int   r0      = rt * 16;
    const int   tid     = threadIdx.x;
    const int   lane    = tid & 31;
    const int   wv      = tid >> 5;          // wave -> 512-column chunk
    const int   half    = lane >> 4;         // C/D rows k (+8 for half 1)
    const int   li      = lane & 15;         // column within tile

    const float2* Qb   = (const float2*)(Q + (size_t)b * NPTS * 2);
    const float*  lwb  = logw + (size_t)b * NPTS;
    const float*  potb = pot  + (size_t)b * NPTS;
    const float*  sqcb = sqc  + (size_t)b * NPTS;

    // cooperative stage: column points + base-2 transformed h into LDS
    for (int j = tid; j < NPTS; j += 128) {
        s_q[j] = Qb[j];
        float hv = lwb[j] - sqcb[j] * inv_eps;
        if (potScale != 0.0f) hv += potb[j] * potScale;
        s_hy[j] = hv * LOG2E;
    }
    __syncthreads();

    // A matrix 16x4 f32: lanes 0-15 = row r0+lane, K=0,1; lanes 16-31 hold
    // K=2,3 == 0 (D padded). Because A's K=2,3 are zero, the B operand never
    // needs masking: finite garbage in its K=2,3 lanes contributes 0.
    const float* Pb = P + (size_t)b * NPTS * 2;
    v2f A;
    if (half == 0) { A.x = Pb[2 * (r0 + li)]; A.y = Pb[2 * (r0 + li) + 1]; }
    else           { A.x = 0.0f;              A.y = 0.0f; }

    const int cBeg = wv * (NPTS / 4);
    const int cEnd = cBeg + (NPTS / 4);

    // ---- pass 1: per-wave partial row maxima over this column chunk ----
    float mrow[8];
#pragma unroll
    for (int k = 0; k < 8; ++k) mrow[k] = -__builtin_inff();

#pragma unroll 4
    for (int c0 = cBeg; c0 < cEnd; c0 += 16) {
        float2 qv  = s_q[c0 + li];
        float  hyv = s_hy[c0 + li];
        v2f Bv; Bv.x = qv.x; Bv.y = qv.y;
        v8f acc = {};
        acc = __builtin_amdgcn_wmma_f32_16x16x4_f32(
            false, A, false, Bv, (short)0, acc, false, false);
#pragma unroll
        for (int k = 0; k < 8; ++k)
            mrow[k] = fmaxf(mrow[k], fmaf(acc[k], ie2, hyv));
    }
    // merge across the 16-lane half (columns) with single-instr swizzles
#pragma unroll
    for (int k = 0; k < 8; ++k) {
        float m = mrow[k];
        m = fmaxf(m, XSWZ(m, SWZ_XOR1));
        m = fmaxf(m, XSWZ(m, SWZ_XOR2));
        m = fmaxf(m, XSWZ(m, SWZ_XOR4));
        m = fmaxf(m, XSWZ(m, SWZ_XOR8));
        mrow[k] = m;
    }
    if (li < 8) red_m[wv][li + 8 * half] = mrow[li];
    __syncthreads();

    // global row maxima (all lanes need M for their 8 rows)
    float Mrow[8];
#pragma unroll
    for (int k = 0; k < 8; ++k) {
        int rk = k + 8 * half;
        Mrow[k] = fmaxf(fmaxf(red_m[0][rk], red_m[1][rk]),
                        fmaxf(red_m[2][rk], red_m[3][rk]));
    }

    // ---- pass 2: partial sums of 2^(arg2 - M) ----
    float srow[8];
#pragma unroll
    for (int k = 0; k < 8; ++k) srow[k] = 0.0f;

#pragma unroll 4
    for (int c0 = cBeg; c0 < cEnd; c0 += 16) {
        float2 qv  = s_q[c0 + li];
        float  hyv = s_hy[c0 + li];
        v2f Bv; Bv.x = qv.x; Bv.y = qv.y;
        v8f acc = {};
        acc = __builtin_amdgcn_wmma_f32_16x16x4_f32(
            false, A, false, Bv, (short)0, acc, false, false);
#pragma unroll
        for (int k = 0; k < 8; ++k) {
            float arg = fmaf(acc[k], ie2, hyv);
            srow[k] += __builtin_amdgcn_exp2f(arg - Mrow[k]);
        }
    }
#pragma unroll
    for (int k = 0; k < 8; ++k) {
        float s = srow[k];
        s += XSWZ(s, SWZ_XOR1);
        s += XSWZ(s, SWZ_XOR2);
        s += XSWZ(s, SWZ_XOR4);
        s += XSWZ(s, SWZ_XOR8);
        srow[k] = s;
    }
    if (li < 8) red_s[wv][li + 8 * half] = srow[li];
    __syncthreads();

    // final combine + writeout: 16 threads, one per row
    if (tid < 16) {
        float M = fmaxf(fmaxf(red_m[0][tid], red_m[1][tid]),
                        fmaxf(red_m[2][tid], red_m[3][tid]));
        float S = (red_s[0][tid] + red_s[1][tid]) +
                  (red_s[2][tid] + red_s[3][tid]);
        size_t idx = (size_t)b * NPTS + r0 + tid;
        out[idx] = (-eps * LN2) * (M + __builtin_amdgcn_logf(S)) + sqr[idx];
    }
}

// -------------------------------------------------------------- update ----
__global__ void update_kernel(float* __restrict__ f, float* __restrict__ g,
                              const float* __restrict__ ft,
                              const float* __restrict__ gt, int n) {
    int i = blockIdx.x * blockDim.x + threadIdx.x;
    if (i >= n) return;
    f[i] = 0.5f * (f[i] + ft[i]);
    g[i] = 0.5f * (g[i] + gt[i]);
}

// ---------------------------------------------------------------- loss ----
__global__ void loss_kernel(const float* __restrict__ a,
                            const float* __restrict__ b,
                            const float* __restrict__ ff,
                            const float* __restrict__ gf,
                            float* __restrict__ out, int n, float invB) {
    __shared__ float red[256];
    float s = 0.0f;
    for (int i = threadIdx.x; i < n; i += 256)
        s += a[i] * ff[i] + b[i] * gf[i];
    red[threadIdx.x] = s;
    __syncthreads();
    for (int off = 128; off > 0; off >>= 1) {
        if (threadIdx.x < off) red[threadIdx.x] += red[threadIdx.x + off];
        __syncthreads();
    }
    if (threadIdx.x == 0) out[0] = red[0] * invB;
}

// -------------------------------------------------------------- driver ----
extern "C" void kernel_launch(void* const* d_in, const int* in_sizes, int n_in,
                              void* d_out, int out_size, void* d_ws,
                              size_t ws_size, hipStream_t stream) {
    (void)in_sizes; (void)n_in; (void)out_size; (void)ws_size;
    const int n = NB * NPTS;

    const float* a = (const float*)d_in[0];
    const float* x = (const float*)d_in[1];
    const float* b = (const float*)d_in[2];
    const float* y = (const float*)d_in[3];

    float* ws   = (float*)d_ws;   // 10 * 16384 floats = 640 KB
    float* la   = ws + 0 * n;
    float* lb   = ws + 1 * n;
    float* sx   = ws + 2 * n;
    float* sy   = ws + 3 * n;
    float* f    = ws + 4 * n;
    float* g    = ws + 5 * n;
    float* ft   = ws + 6 * n;
    float* gt   = ws + 7 * n;
    float* ffin = ws + 8 * n;
    float* gfin = ws + 9 * n;

    prep_kernel<<<(n + 255) / 256, 256, 0, stream>>>(a, x, b, y, la, lb, sx,
                                                     sy, n);

    // geomloss annealing schedule: eps = diameter^p .. blur^p, factor 0.64
    double epsList[64];
    int ne = 0;
    const double eps_final = 0.05 * 0.05;
    double e = 2.0 * 2.0;
    while (e > eps_final && ne < 63) { epsList[ne++] = e; e *= 0.8 * 0.8; }
    epsList[ne++] = eps_final;

    const int SGRID = NB * 128;  // 1024 workgroups of 128 threads

    // init at largest temperature (potentials unused: potScale = 0)
    float e0 = (float)epsList[0];
    softmin_kernel<<<SGRID, 128, 0, stream>>>(x, y, lb, g, sx, sy, f, e0, 0.f);
    softmin_kernel<<<SGRID, 128, 0, stream>>>(y, x, la, f, sy, sx, g, e0, 0.f);

    // annealed symmetric (damped) Sinkhorn loop
    for (int it = 0; it < ne; ++it) {
        float ei  = (float)epsList[it];
        float inv = 1.0f / ei;
        softmin_kernel<<<SGRID, 128, 0, stream>>>(x, y, lb, g, sx, sy, ft, ei,
                                                  inv);
        softmin_kernel<<<SGRID, 128, 0, stream>>>(y, x, la, f, sy, sx, gt, ei,
                                                  inv);
        update_kernel<<<(n + 255) / 256, 256, 0, stream>>>(f, g, ft, gt, n);
    }

    // final extrapolation at blur temperature
    float ef   = (float)epsList[ne - 1];
    float invf = 1.0f / ef;
    softmin_kernel<<<SGRID, 128, 0, stream>>>(x, y, lb, g, sx, sy, ffin, ef,
                                              invf);
    softmin_kernel<<<SGRID, 128, 0, stream>>>(y, x, la, f, sy, sx, gfin, ef,
                                              invf);

    loss_kernel<<<1, 256, 0, stream>>>(a, b, ffin, gfin, (float*)d_out, n,
                                       1.0f / (float)NB);
}